// HungarianMatcher_21569325760768
// MI455X (gfx1250) — compile-verified
//
#include <hip/hip_runtime.h>
#include <hip/hip_bf16.h>
#include <math.h>

typedef __attribute__((ext_vector_type(2))) float v2f;
typedef __attribute__((ext_vector_type(8))) float v8f;

#define KMAX            512   // max (N + M) supported by solver LDS state
#define SOLVER_THREADS  512   // 16 wave32 waves on one WGP
#define SOLVER_WAVES    (SOLVER_THREADS / 32)

// ---------------------------------------------------------------------------
// Kernel 1: union-base outer sum via the matrix core.
//   Ubase[i][j] = area_a[i] + area_b[j]
// expressed as a rank-2 GEMM  [area_a | 1] (16x4, K slots 0,1)  x
//                             [ones ; area_b] (4x16)  ->  16x16 f32 tile,
// one V_WMMA_F32_16X16X4_F32 per wave.
// ---------------------------------------------------------------------------
__global__ __launch_bounds__(32)
void hm_union_base_wmma(const float* __restrict__ tgt,
                        const float* __restrict__ prop,
                        float* __restrict__ ubase, int N, int M) {
  const int tilesM = (M + 15) >> 4;
  const int tile   = blockIdx.x;
  const int ti     = tile / tilesM;
  const int tj     = tile % tilesM;
  const int lane   = threadIdx.x;   // 0..31
  const int half   = lane >> 4;     // 0: K0/K1 regs, 1: K2/K3 regs (zero)
  const int l15    = lane & 15;

  v2f a = {0.0f, 0.0f};   // A 16x4: vgpr0=K0, vgpr1=K1 for lanes 0-15
  v2f b = {0.0f, 0.0f};   // B 4x16: vgpr0=row K0, vgpr1=row K1 for lanes 0-15
  if (half == 0) {
    const int gi = ti * 16 + l15;
    if (gi < N) {
      const float x0 = tgt[gi * 4 + 0], y0 = tgt[gi * 4 + 1];
      const float x1 = tgt[gi * 4 + 2], y1 = tgt[gi * 4 + 3];
      a.x = (x1 - x0) * (y1 - y0);  // K=0 column: area_a[m]
      a.y = 1.0f;                   // K=1 column: ones
    }
    const int gj = tj * 16 + l15;
    if (gj < M) {
      const float x0 = prop[gj * 4 + 0], y0 = prop[gj * 4 + 1];
      const float x1 = prop[gj * 4 + 2], y1 = prop[gj * 4 + 3];
      b.x = 1.0f;                   // K=0 row: ones
      b.y = (x1 - x0) * (y1 - y0);  // K=1 row: area_b[n]
    }
  }

  v8f c = {};
  // D = A x B + 0 : D[m][n] = area_a[m] + area_b[n]
  c = __builtin_amdgcn_wmma_f32_16x16x4_f32(
      /*neg_a=*/false, a, /*neg_b=*/false, b,
      /*c_mod=*/(short)0, c, /*reuse_a=*/false, /*reuse_b=*/false);

  // D layout: VGPR r, lanes 0-15 -> M=r, N=lane; lanes 16-31 -> M=8+r, N=lane-16
  const int mbase = half ? 8 : 0;
#pragma unroll
  for (int r = 0; r < 8; ++r) {
    const int gi = ti * 16 + mbase + r;
    const int gj = tj * 16 + l15;
    if (gi < N && gj < M) ubase[gi * M + gj] = c[r];
  }
}

// ---------------------------------------------------------------------------
// Kernel 2: finish IoU cost in f32 (matching jnp float32 math), widen to f64.
//   cost[i][j] = -( inter / (Ubase - inter) )
// ---------------------------------------------------------------------------
__global__ __launch_bounds__(256)
void hm_cost_kernel(const float* __restrict__ tgt, const float* __restrict__ prop,
                    const float* __restrict__ ubase, double* __restrict__ Cmat,
                    int N, int M) {
  const int idx = blockIdx.x * blockDim.x + threadIdx.x;
  if (idx >= N * M) return;
  const int i = idx / M;
  const int j = idx - i * M;
  const float ax0 = tgt[i * 4 + 0], ay0 = tgt[i * 4 + 1];
  const float ax1 = tgt[i * 4 + 2], ay1 = tgt[i * 4 + 3];
  const float bx0 = prop[j * 4 + 0], by0 = prop[j * 4 + 1];
  const float bx1 = prop[j * 4 + 2], by1 = prop[j * 4 + 3];
  float w = fminf(ax1, bx1) - fmaxf(ax0, bx0);
  float h = fminf(ay1, by1) - fmaxf(ay0, by0);
  w = fmaxf(w, 0.0f);
  h = fmaxf(h, 0.0f);
  const float inter = w * h;
  const float uni   = ubase[idx] - inter;   // (area_a + area_b) - inter
  const float iou   = inter / uni;
  Cmat[idx] = -(double)iou;
}

// ---------------------------------------------------------------------------
// Kernel 3: exact Jonker-Volgenant LAP on the K=N+M extended matrix,
// one workgroup, all state in LDS. Extended cost Ce(i,j) is generated
// on the fly: C[i][j] in the real block, pad (= cost_limit/2) in the
// row/col dummy strips, 0 in the dummy-dummy block.
// ---------------------------------------------------------------------------
__global__ __launch_bounds__(SOLVER_THREADS)
void hm_lapjv_kernel(const double* __restrict__ Cmat, int N, int M,
                     double pad, int* __restrict__ outv) {
  const int K    = N + M;
  const int tid  = threadIdx.x;
  const int lane = tid & 31;
  const int wid  = tid >> 5;

  __shared__ double s_spc[KMAX];
  __shared__ double s_u[KMAX];
  __shared__ double s_v[KMAX];
  __shared__ int    s_path[KMAX];
  __shared__ int    s_SC[KMAX];      // scanned columns
  __shared__ int    s_SR[KMAX];      // scanned rows
  __shared__ int    s_col4row[KMAX];
  __shared__ int    s_row4col[KMAX];
  __shared__ double s_rmin[SOLVER_WAVES];
  __shared__ int    s_ridx[SOLVER_WAVES];
  __shared__ int    s_i, s_sink;
  __shared__ double s_minVal;

  const double INF = 1e300;

  for (int j = tid; j < K; j += SOLVER_THREADS) {
    s_u[j] = 0.0;
    s_v[j] = 0.0;
    s_col4row[j] = -1;
    s_row4col[j] = -1;
  }
  __syncthreads();

  for (int cur = 0; cur < K; ++cur) {
    // per-row reset
    for (int j = tid; j < K; j += SOLVER_THREADS) {
      s_spc[j]  = INF;
      s_path[j] = -1;
      s_SC[j]   = 0;
      s_SR[j]   = 0;
    }
    if (tid == 0) { s_i = cur; s_sink = -1; s_minVal = 0.0; }
    __syncthreads();

    // ---- Dijkstra shortest augmenting path ----
    for (;;) {
      if (s_sink != -1) break;     // consistent: body ends with __syncthreads()
      const int    irow = s_i;
      const double mv   = s_minVal;
      const double ui   = s_u[irow];
      if (tid == 0) s_SR[irow] = 1;  // SR.append(i)

      // relax all unscanned columns (each j owned by exactly one thread)
      for (int j = tid; j < K; j += SOLVER_THREADS) {
        if (!s_SC[j]) {
          double cij;
          if (irow < N) cij = (j < M) ? Cmat[irow * M + j] : pad;
          else          cij = (j < M) ? pad : 0.0;
          const double red = mv + cij - ui - s_v[j];
          if (red < s_spc[j]) { s_spc[j] = red; s_path[j] = irow; }
        }
      }
      __syncthreads();

      // argmin over unscanned columns; ties -> smallest j (numpy argmin order)
      double bv = INF;
      int    bj = 0x7fffffff;
      for (int j = tid; j < K; j += SOLVER_THREADS) {
        if (!s_SC[j]) {
          const double sv = s_spc[j];
          if (sv < bv || (sv == bv && j < bj)) { bv = sv; bj = j; }
        }
      }
#pragma unroll
      for (int off = 16; off > 0; off >>= 1) {
        const double ov = __shfl_down(bv, off);
        const int    oj = __shfl_down(bj, off);
        if (ov < bv || (ov == bv && oj < bj)) { bv = ov; bj = oj; }
      }
      if (lane == 0) { s_rmin[wid] = bv; s_ridx[wid] = bj; }
      __syncthreads();

      if (tid == 0) {
        double fbv = s_rmin[0];
        int    fbj = s_ridx[0];
        for (int w = 1; w < SOLVER_WAVES; ++w) {
          const double ov = s_rmin[w];
          const int    oj = s_ridx[w];
          if (ov < fbv || (ov == fbv && oj < fbj)) { fbv = ov; fbj = oj; }
        }
        s_minVal  = fbv;
        s_SC[fbj] = 1;
        const int r = s_row4col[fbj];
        if (r == -1) s_sink = fbj;
        else         s_i = r;
      }
      __syncthreads();
    }

    // ---- dual variable updates ----
    const double mv = s_minVal;
    for (int j = tid; j < K; j += SOLVER_THREADS) {
      if (s_SR[j]) {
        if (j == cur) s_u[j] += mv;
        else          s_u[j] += mv - s_spc[s_col4row[j]];
      }
      if (s_SC[j]) s_v[j] -= mv - s_spc[j];
    }
    __syncthreads();

    // ---- augment along the found path (sequential chain) ----
    if (tid == 0) {
      int j = s_sink;
      for (;;) {
        const int i  = s_path[j];
        s_row4col[j] = i;
        const int nj = s_col4row[i];
        s_col4row[i] = j;
        j = nj;
        if (i == cur) break;
      }
    }
    __syncthreads();
  }

  // ---- emit (row[N], col[M]) with dummy matches mapped to -1 ----
  for (int i = tid; i < N; i += SOLVER_THREADS) {
    const int r = s_col4row[i];
    outv[i] = (r >= M) ? -1 : r;
  }
  for (int j = tid; j < M; j += SOLVER_THREADS) {
    const int c = s_row4col[j];
    outv[N + j] = (c >= N) ? -1 : c;
  }
}

// ---------------------------------------------------------------------------
extern "C" void kernel_launch(void* const* d_in, const int* in_sizes, int n_in,
                              void* d_out, int out_size, void* d_ws, size_t ws_size,
                              hipStream_t stream) {
  const float* tgt  = (const float*)d_in[0];   // [N,4] f32
  const float* prop = (const float*)d_in[1];   // [M,4] f32
  const int N = in_sizes[0] / 4;               // 100
  const int M = in_sizes[1] / 4;               // 300

  // workspace: [ Cmat f64 NxM | ubase f32 NxM ]
  double* Cmat  = (double*)d_ws;
  float*  ubase = (float*)((char*)d_ws + (size_t)N * (size_t)M * sizeof(double));

  const int tilesN = (N + 15) / 16;
  const int tilesM = (M + 15) / 16;
  hipLaunchKernelGGL(hm_union_base_wmma, dim3(tilesN * tilesM), dim3(32), 0, stream,
                     tgt, prop, ubase, N, M);

  const int total = N * M;
  hipLaunchKernelGGL(hm_cost_kernel, dim3((total + 255) / 256), dim3(256), 0, stream,
                     tgt, prop, ubase, Cmat, N, M);

  // cost_limit = -IOU_THRESHOLD = -0.5  ->  padding = cost_limit / 2 = -0.25
  hipLaunchKernelGGL(hm_lapjv_kernel, dim3(1), dim3(SOLVER_THREADS), 0, stream,
                     Cmat, N, M, -0.25, (int*)d_out);
}